// BasicLayer_63995012710460
// MI455X (gfx1250) — compile-verified
//
#include <hip/hip_runtime.h>
#include <hip/hip_bf16.h>
#include <math.h>

// ---------------- problem constants (from reference) ----------------
#define DIMC   128
#define HEADS  4
#define DHD    32
#define KSZ    7
#define HFF    340          // int(128 * 2.66)
#define HWD    96
#define NPIX   (HWD * HWD)  // 9216
#define NDEPTH 2
#define RPBW   (2 * KSZ - 1) // 13

typedef __attribute__((ext_vector_type(2))) float v2f;
typedef __attribute__((ext_vector_type(8))) float v8f;

// ---------------- wave32 helpers ----------------
__device__ __forceinline__ float wave_reduce_sum(float v) {
    v += __shfl_xor(v, 16, 32);
    v += __shfl_xor(v, 8, 32);
    v += __shfl_xor(v, 4, 32);
    v += __shfl_xor(v, 2, 32);
    v += __shfl_xor(v, 1, 32);
    return v;
}

// ---------------- LayerNorm: one wave per 128-channel row ----------------
__global__ void ln_kernel(const float* __restrict__ x,
                          const float* __restrict__ g,
                          const float* __restrict__ b,
                          float* __restrict__ y) {
    const int wave = threadIdx.x >> 5;
    const int lane = threadIdx.x & 31;
    const int row  = blockIdx.x * 8 + wave;   // grid sized so row < NPIX
    const float4 xv = *(const float4*)(x + (size_t)row * DIMC + lane * 4);
    float s = xv.x + xv.y + xv.z + xv.w;
    const float m = wave_reduce_sum(s) * (1.0f / DIMC);
    const float dx0 = xv.x - m, dx1 = xv.y - m, dx2 = xv.z - m, dx3 = xv.w - m;
    float vs = dx0 * dx0 + dx1 * dx1 + dx2 * dx2 + dx3 * dx3;
    const float var = wave_reduce_sum(vs) * (1.0f / DIMC);
    const float inv = rsqrtf(var + 1e-6f);
    const float4 gv = *(const float4*)(g + lane * 4);
    const float4 bv = *(const float4*)(b + lane * 4);
    float4 yv;
    yv.x = dx0 * inv * gv.x + bv.x;
    yv.y = dx1 * inv * gv.y + bv.y;
    yv.z = dx2 * inv * gv.z + bv.z;
    yv.w = dx3 * inv * gv.w + bv.w;
    *(float4*)(y + (size_t)row * DIMC + lane * 4) = yv;
}

// ---------------- fp32 WMMA GEMM: C[M,N] = A[M,K] * W[N,K]^T (+bias)(+add) ----------------
// One wave computes one 16x16 tile of C via V_WMMA_F32_16X16X4_F32.
// f32 A fragment (16x4): lanes 0-15 rows M, VGPR pair = K {k,k+1}; lanes 16-31 = K {k+2,k+3}.
// f32 B fragment (4x16): same K-pair striping; N striped across lanes within half-wave.
// Both fragments are contiguous K-pairs per lane -> single b64 loads.
//
// N-edge handling: out-of-range B columns load from a CLAMPED valid weight row.
// WMMA column n only feeds C column n, and the store guard drops cols >= N, so
// the garbage lanes are never observed. This keeps EXEC all-1s (WMMA
// requirement) with zero divergence in the hot loop.
__global__ void gemm_wmma_f32(const float* __restrict__ A,
                              const float* __restrict__ W,
                              const float* __restrict__ bias,   // may be null
                              const float* __restrict__ addsrc, // may be null (residual, ld = N)
                              float* __restrict__ C,
                              int M, int N, int K) {
    const int wave = threadIdx.x >> 5;
    const int lane = threadIdx.x & 31;
    const int half = lane >> 4;   // 0 or 1: which K-pair
    const int r    = lane & 15;   // row (A) / col (B,C) within tile
    const int m0   = blockIdx.y * 16;
    const int n0   = (blockIdx.x * 4 + wave) * 16;

    const int   ncl  = min(n0 + r, N - 1);       // clamped weight row
    const float* arow = A + (size_t)(m0 + r) * K + 2 * half;
    const float* wrow = W + (size_t)ncl * K + 2 * half;

    v8f acc = {};
#pragma unroll 4
    for (int k = 0; k < K; k += 4) {
        __builtin_prefetch(wrow + k + 64, 0, 3);  // global_prefetch_b8, RT
        const v2f a = *(const v2f*)(arow + k);    // global_load_b64
        const v2f b = *(const v2f*)(wrow + k);    // global_load_b64
        acc = __builtin_amdgcn_wmma_f32_16x16x4_f32(
            /*neg_a=*/false, a, /*neg_b=*/false, b,
            /*c_mod=*/(short)0, acc, /*reuse_a=*/false, /*reuse_b=*/false);
    }

    const int col = n0 + r;
    if (col < N) {
        const float bb = bias ? bias[col] : 0.0f;
#pragma unroll
        for (int v = 0; v < 8; ++v) {
            const int row = m0 + v + 8 * half;  // C/D tile: VGPR v -> M = v (+8 for hi half)
            float o = acc[v] + bb;
            if (addsrc) o += addsrc[(size_t)row * N + col];
            C[(size_t)row * N + col] = o;
        }
    }
}

// ---------------- 7x7 neighborhood attention: block=pixel, wave=head, lane=d ----------------
__global__ void na2d_kernel(const float* __restrict__ qkv,  // (NPIX, 3*DIMC)
                            const float* __restrict__ rpb,  // (HEADS, 13, 13)
                            float* __restrict__ out) {      // (NPIX, DIMC)
    const int h    = threadIdx.x >> 5;
    const int lane = threadIdx.x & 31;
    const int p    = blockIdx.x;
    const int i = p / HWD, j = p % HWD;
    const int si = min(max(i - 3, 0), HWD - KSZ);
    const int sj = min(max(j - 3, 0), HWD - KSZ);

    const float q = qkv[(size_t)p * (3 * DIMC) + h * DHD + lane] * 0.17677669529663687f; // 32^-0.5

    float logit[KSZ * KSZ];
    float mx = -3.4e38f;
#pragma unroll
    for (int a = 0; a < KSZ; ++a) {
        const int ih = si + a;
        const int rh = ih - i + (KSZ - 1);
#pragma unroll
        for (int c = 0; c < KSZ; ++c) {
            const int iw = sj + c;
            const int np = ih * HWD + iw;
            const float kv = qkv[(size_t)np * (3 * DIMC) + DIMC + h * DHD + lane];
            float s = wave_reduce_sum(q * kv);  // all lanes get full dot product
            const int rw = iw - j + (KSZ - 1);
            s += rpb[(h * RPBW + rh) * RPBW + rw];
            logit[a * KSZ + c] = s;
            mx = fmaxf(mx, s);
        }
    }
    float denom = 0.0f;
#pragma unroll
    for (int n = 0; n < KSZ * KSZ; ++n) {
        logit[n] = expf(logit[n] - mx);
        denom += logit[n];
    }
    const float inv = 1.0f / denom;

    float o = 0.0f;
#pragma unroll
    for (int a = 0; a < KSZ; ++a) {
        const int ih = si + a;
#pragma unroll
        for (int c = 0; c < KSZ; ++c) {
            const int iw = sj + c;
            const int np = ih * HWD + iw;
            o += logit[a * KSZ + c] * qkv[(size_t)np * (3 * DIMC) + 2 * DIMC + h * DHD + lane];
        }
    }
    out[(size_t)p * DIMC + h * DHD + lane] = o * inv;
}

// ---------------- depthwise 3x3 conv + bias + exact-GELU gate ----------------
// in: h (NPIX, 2*HFF); out: g (NPIX, HFF) = gelu(dw(h)[:HFF]) * dw(h)[HFF:]
__global__ void dw_gelu_kernel(const float* __restrict__ hbuf,
                               const float* __restrict__ w,    // (3,3,1,2*HFF)
                               const float* __restrict__ bias, // (2*HFF)
                               float* __restrict__ g) {
    const int idx = blockIdx.x * blockDim.x + threadIdx.x;
    if (idx >= NPIX * HFF) return;
    const int c = idx % HFF;
    const int p = idx / HFF;
    const int i = p / HWD, j = p % HWD;
    float s1 = bias[c];
    float s2 = bias[c + HFF];
#pragma unroll
    for (int di = 0; di < 3; ++di) {
        const int ii = i + di - 1;
        if (ii < 0 || ii >= HWD) continue;
#pragma unroll
        for (int dj = 0; dj < 3; ++dj) {
            const int jj = j + dj - 1;
            if (jj < 0 || jj >= HWD) continue;
            const float* hp = hbuf + (size_t)(ii * HWD + jj) * (2 * HFF);
            const float* wp = w + (size_t)(di * 3 + dj) * (2 * HFF);
            s1 += hp[c]       * wp[c];
            s2 += hp[c + HFF] * wp[c + HFF];
        }
    }
    const float ge = 0.5f * s1 * (1.0f + erff(s1 * 0.7071067811865476f));
    g[(size_t)p * HFF + c] = ge * s2;
}

// ---------------- host-side orchestration ----------------
extern "C" void kernel_launch(void* const* d_in, const int* in_sizes, int n_in,
                              void* d_out, int out_size, void* d_ws, size_t ws_size,
                              hipStream_t stream) {
    (void)in_sizes; (void)n_in; (void)out_size; (void)ws_size;

    const float* x_in     = (const float*)d_in[0];
    const float* norm1_g  = (const float*)d_in[1];
    const float* norm1_b  = (const float*)d_in[2];
    const float* qkv_w    = (const float*)d_in[3];
    const float* qkv_b    = (const float*)d_in[4];
    const float* rpb      = (const float*)d_in[5];
    const float* proj_w   = (const float*)d_in[6];
    const float* proj_b   = (const float*)d_in[7];
    const float* norm2_g  = (const float*)d_in[8];
    const float* norm2_b  = (const float*)d_in[9];
    const float* ffn_in_w = (const float*)d_in[10];
    const float* ffn_dw_w = (const float*)d_in[11];
    const float* ffn_dw_b = (const float*)d_in[12];
    const float* ffn_out_w= (const float*)d_in[13];

    float* ws = (float*)d_ws;
    float* y    = ws;                         // NPIX*DIMC
    float* qkvb = y    + (size_t)NPIX * DIMC; // NPIX*3*DIMC
    float* ab   = qkvb + (size_t)NPIX * 3 * DIMC;
    float* x1   = ab   + (size_t)NPIX * DIMC; // after attention residual
    float* x2   = x1   + (size_t)NPIX * DIMC; // layer output ping-pong
    float* hb   = x2   + (size_t)NPIX * DIMC; // NPIX*2*HFF
    float* gb   = hb   + (size_t)NPIX * 2 * HFF; // NPIX*HFF

    const dim3 blk128(128);
    const dim3 ln_grid(NPIX / 8);
    const dim3 blk256(256);

    for (int l = 0; l < NDEPTH; ++l) {
        const float* xin = (l == 0) ? x_in : x2;
        const float* n1g = norm1_g + l * DIMC;
        const float* n1b = norm1_b + l * DIMC;
        const float* qw  = qkv_w   + (size_t)l * 3 * DIMC * DIMC;
        const float* qb  = qkv_b   + (size_t)l * 3 * DIMC;
        const float* rp  = rpb     + (size_t)l * HEADS * RPBW * RPBW;
        const float* pw  = proj_w  + (size_t)l * DIMC * DIMC;
        const float* pb  = proj_b  + (size_t)l * DIMC;
        const float* n2g = norm2_g + l * DIMC;
        const float* n2b = norm2_b + l * DIMC;
        const float* fiw = ffn_in_w + (size_t)l * 2 * HFF * DIMC;
        const float* fdw = ffn_dw_w + (size_t)l * 9 * 2 * HFF;
        const float* fdb = ffn_dw_b + (size_t)l * 2 * HFF;
        const float* fow = ffn_out_w + (size_t)l * DIMC * HFF;

        // y = LN1(x)
        ln_kernel<<<ln_grid, blk256, 0, stream>>>(xin, n1g, n1b, y);
        // qkv = y * qkv_w^T + qkv_b        (M=9216, N=384, K=128)
        gemm_wmma_f32<<<dim3(384 / 64, NPIX / 16), blk128, 0, stream>>>(
            y, qw, qb, nullptr, qkvb, NPIX, 3 * DIMC, DIMC);
        // a = NA2D(q,k,v) + rpb
        na2d_kernel<<<dim3(NPIX), blk128, 0, stream>>>(qkvb, rp, ab);
        // x1 = a * proj_w^T + proj_b + x   (M=9216, N=128, K=128)
        gemm_wmma_f32<<<dim3(128 / 64, NPIX / 16), blk128, 0, stream>>>(
            ab, pw, pb, xin, x1, NPIX, DIMC, DIMC);
        // y = LN2(x1)
        ln_kernel<<<ln_grid, blk256, 0, stream>>>(x1, n2g, n2b, y);
        // h = y * ffn_in_w^T               (M=9216, N=680, K=128) -- N edge handled
        gemm_wmma_f32<<<dim3((2 * HFF + 63) / 64, NPIX / 16), blk128, 0, stream>>>(
            y, fiw, nullptr, nullptr, hb, NPIX, 2 * HFF, DIMC);
        // g = gelu(dw(h)+b)[:HFF] * (dw(h)+b)[HFF:]
        dw_gelu_kernel<<<dim3((NPIX * HFF + 255) / 256), blk256, 0, stream>>>(
            hb, fdw, fdb, gb);
        // xout = g * ffn_out_w^T + x1      (M=9216, N=128, K=340)
        float* xout = (l == NDEPTH - 1) ? (float*)d_out : x2;
        gemm_wmma_f32<<<dim3(128 / 64, NPIX / 16), blk128, 0, stream>>>(
            gb, fow, nullptr, x1, xout, NPIX, DIMC, HFF);
    }
}